// BistableSR_67456756351641
// MI455X (gfx1250) — compile-verified
//
#include <hip/hip_runtime.h>
#include <stdint.h>

// Bistable SR, RK4 over T=4096 steps, 4096 independent trajectories (64x64),
// layout (B, C, T) with T contiguous. One trajectory per lane; inputs staged
// through LDS with CDNA5 async-to-LDS loads (ASYNCcnt), double-buffered.

#define T_TOTAL   4096
#define TILE      64                   // time steps per LDS tile
#define NROWS     32                   // trajectories per block (1 wave32)
#define LSTRIDE   68                   // floats per LDS row: 64 + 4 pad (272B, 16B aligned)
#define NTILES    (T_TOTAL / TILE)     // 64
#define BUF_FLTS  (NROWS * LSTRIDE)    // 2176 floats = 8704 B per buffer
#define CHUNKS    (TILE / 4)           // float4 chunks per row per tile

typedef float v4f __attribute__((ext_vector_type(4)));

template <int N>
__device__ __forceinline__ void wait_asynccnt() {
#if __has_builtin(__builtin_amdgcn_s_wait_asynccnt)
  __builtin_amdgcn_s_wait_asynccnt(N);
#else
  asm volatile("s_wait_asynccnt %0" ::"i"(N) : "memory");
#endif
}

// CDNA5 async copy: copies 16 bytes per lane from global base plus offset goff to LDS address lds_off.
// Tracked by ASYNCcnt (in-order completion among async loads).
__device__ __forceinline__ void async_load_b128(uint32_t lds_off, uint32_t goff,
                                                const float* base) {
  asm volatile("global_load_async_to_lds_b128 %0, %1, %2"
               ::"v"(lds_off), "v"(goff), "s"(base)
               : "memory");
}

// One RK4 step of dy/dt = a*y - b*y^3 + inp, plus noise * sqrt(dt).
// Evaluation order mirrors the JAX reference.
__device__ __forceinline__ float rk4_step(float y, float inp, float nn, float sdt) {
  const float a = 1.0f, b = 1.0f, inten = 1.0f;
  const float dt  = 0.01f;
  const float hdt = 0.5f * dt;
  const float dt6 = dt / 6.0f;
  float k1 = a * y - b * (y * y * y) + inp;
  float y2 = y + hdt * k1;
  float k2 = a * y2 - b * (y2 * y2 * y2) + inp;
  float y3 = y + hdt * k2;
  float k3 = a * y3 - b * (y3 * y3 * y3) + inp;
  float y4 = y + dt * k3;
  float k4 = a * y4 - b * (y4 * y4 * y4) + inp;
  return y + dt6 * (k1 + 2.0f * k2 + 2.0f * k3 + k4) + inten * nn * sdt;
}

__global__ void __launch_bounds__(NROWS)
bistable_rk4(const float* __restrict__ x, const float* __restrict__ nz,
             float* __restrict__ out) {
  extern __shared__ __align__(16) float smem[];   // [2][x]: 2 bufs, then [2][n]
  float* xs = smem;                  // xs buffer p at smem + p*BUF_FLTS
  float* ns = smem + 2 * BUF_FLTS;   // ns buffer p at ns   + p*BUF_FLTS

  const int tid  = threadIdx.x;              // 0..31
  const int row0 = blockIdx.x * NROWS;       // first trajectory of this block

  // Cooperative load mapping: lane covers (row = 2*i + sub, cols c4..c4+3).
  const int sub = tid >> 4;                  // 0 or 1
  const int c4  = (tid & 15) << 2;           // float column 0,4,...,60

  const uint32_t xs_base  = (uint32_t)(uintptr_t)xs;   // LDS byte offsets
  const uint32_t ns_base  = (uint32_t)(uintptr_t)ns;
  const uint32_t lane_lds = (uint32_t)((sub * LSTRIDE + c4) * 4);
  const uint32_t g_lane0  = (uint32_t)(((row0 + sub) * T_TOTAL + c4) * 4);

  auto issue_tile = [&](int k, int p) {
    uint32_t g  = g_lane0 + (uint32_t)(k * TILE * 4);
    uint32_t lx = xs_base + (uint32_t)p * (BUF_FLTS * 4) + lane_lds;
    uint32_t ln = ns_base + (uint32_t)p * (BUF_FLTS * 4) + lane_lds;
#pragma unroll
    for (int i = 0; i < 16; ++i) {           // 2 rows per instr * 16 = 32 rows
      async_load_b128(lx, g, x);
      async_load_b128(ln, g, nz);
      g  += 2 * T_TOTAL * 4;                 // advance 2 trajectory rows
      lx += 2 * LSTRIDE * 4;
      ln += 2 * LSTRIDE * 4;
    }
  };

  const float sdt = sqrtf(0.01f);            // folds to nearest f32 (0.1f)
  float y = 0.0f;
  float* outrow = out + (size_t)(row0 + tid) * T_TOTAL;

  issue_tile(0, 0);                          // prefetch tile 0
#pragma unroll 1
  for (int k = 0; k < NTILES; ++k) {
    const int p = k & 1;
    if (k + 1 < NTILES) {
      issue_tile(k + 1, p ^ 1);              // prefetch next tile
      wait_asynccnt<32>();                   // only newest 32 (tile k+1) remain
    } else {
      wait_asynccnt<0>();
    }
    __syncthreads();                         // single-wave WG: effectively free

    const v4f* xr = (const v4f*)(xs + p * BUF_FLTS + tid * LSTRIDE);
    const v4f* nr = (const v4f*)(ns + p * BUF_FLTS + tid * LSTRIDE);
    v4f* orow = (v4f*)(outrow + k * TILE);

#pragma unroll 4
    for (int t4 = 0; t4 < CHUNKS; ++t4) {
      v4f xi = xr[t4];                       // ds_load_b128, padded stride
      v4f ni = nr[t4];
      v4f o;
      y = o.x = rk4_step(y, xi.x, ni.x, sdt);
      y = o.y = rk4_step(y, xi.y, ni.y, sdt);
      y = o.z = rk4_step(y, xi.z, ni.z, sdt);
      y = o.w = rk4_step(y, xi.w, ni.w, sdt);
      __builtin_nontemporal_store(o, orow + t4);   // write-once: bypass-ish
    }
    __syncthreads();                         // protect buf before re-fill
  }
}

extern "C" void kernel_launch(void* const* d_in, const int* in_sizes, int n_in,
                              void* d_out, int out_size, void* d_ws, size_t ws_size,
                              hipStream_t stream) {
  (void)in_sizes; (void)n_in; (void)out_size; (void)d_ws; (void)ws_size;
  const float* x  = (const float*)d_in[0];
  const float* nz = (const float*)d_in[1];
  float* out = (float*)d_out;

  const int ntraj = 64 * 64;                 // 4096 trajectories
  dim3 grid(ntraj / NROWS);                  // 128 blocks
  dim3 block(NROWS);                         // 32 threads = 1 wave32
  size_t shmem = (size_t)4 * BUF_FLTS * sizeof(float);  // 34816 B
  hipLaunchKernelGGL(bistable_rk4, grid, block, shmem, stream, x, nz, out);
}